// SSIMLoss_19052474925663
// MI455X (gfx1250) — compile-verified
//
#include <hip/hip_runtime.h>
#include <math.h>

// CDNA5 (gfx1250) SSIM loss. Separable 11x11 Gaussian blur routed through the
// fp32 matrix pipe: V_WMMA_F32_16X16X4_F32 computes banded-matrix products
//   Out = In(16x28) x Wband(28x16)   (horizontal blur)
//   Out = Wband(16x28) x Hb(28x16)   (vertical blur)
// in full fp32 precision (required: sigma = E[x^2] - mu^2 is a cancellation).
//
// Round-3 changes: Hb staged TRANSPOSED in LDS so each C/D fragment spills as
// two ds_store_b128 (was eight ds_store_b32); row blocks at bases {0,12} keep
// 16B alignment and make the K-pad rows (26,27) carry real finite data that the
// zero band-weights annihilate — no explicit zero-padding pass needed.

typedef __attribute__((ext_vector_type(2))) float v2f;
typedef __attribute__((ext_vector_type(8))) float v8f;

#define WSZ    11
#define RAD    5
#define TILE_H 32
#define TILE_W 64
#define IN_H   44                   // 32 + 10 halo + 2 rows feeding zero-weight K-pad
#define IN_W   76                   // 64 + 10 halo, padded to 76 (covers K-pad reads)
#define HB_R   28                   // 26 halo rows + 2 zero-weight K-pad rows
#define GPAD   15                   // weight index offset: d in [-15, 28] -> [0, 43]
#define GLEN   44
#define NWAVE  8

__global__ __launch_bounds__(256)
void ssim_main_kernel(const float* __restrict__ pred,
                      const float* __restrict__ tgt,
                      const float* __restrict__ window,
                      float* __restrict__ partial,
                      int H, int W) {
    __shared__ float sP[IN_H][IN_W];
    __shared__ float sT[IN_H][IN_W];
    __shared__ __align__(16) float sHbT[NWAVE][16][HB_R];  // [col][row], col-major Hb
    __shared__ float sGpad[GLEN];   // zeros except [GPAD..GPAD+10] = g[0..10]
    __shared__ float sRed[NWAVE];

    const int tid = threadIdx.x;
    const int bz  = blockIdx.z;
    const int wgy = blockIdx.y * TILE_H;
    const int wgx = blockIdx.x * TILE_W;
    const float* pb = pred + (size_t)bz * H * W;
    const float* tb = tgt  + (size_t)bz * H * W;

    // Recover the 1-D Gaussian from the 2-D window input:
    // window[5][j] = g[5]*g[j]  =>  g[j] = window[5][j] / sqrt(window[5][5]).
    // Stored zero-padded so banded-weight gathers need no bounds check.
    if (tid == 0) {
        for (int j = 0; j < GLEN; ++j) sGpad[j] = 0.0f;
        float g5 = sqrtf(window[5 * WSZ + 5]);
        for (int j = 0; j < WSZ; ++j) sGpad[GPAD + j] = window[5 * WSZ + j] / g5;
    }

    // Cooperative stage of pred/target (+halo) with zero SAME-padding.
    for (int i = tid; i < IN_H * IN_W; i += 256) {
        int r = i / IN_W, c = i % IN_W;
        int gy = wgy - RAD + r, gx = wgx - RAD + c;
        bool ok = (gy >= 0) && (gy < H) && (gx >= 0) && (gx < W);
        size_t idx = (size_t)gy * W + gx;
        sP[r][c] = ok ? pb[idx] : 0.0f;
        sT[r][c] = ok ? tb[idx] : 0.0f;
    }

    const int wave = tid >> 5;       // wave32
    const int lane = tid & 31;
    const int half = lane >> 4;      // 0: lanes 0-15, 1: lanes 16-31
    const int l16  = lane & 15;
    const int ty   = (wave >> 2) * 16;   // wave subtile within the 32x64 block tile
    const int tx   = (wave & 3) * 16;

    __syncthreads();

    // Weight fragments, shared by H-blur B-matrix and V-blur A-matrix:
    // per lane, step s: { g[k - l16], g[k+1 - l16] } with k = 4s + 2*half.
    // Unconditional LDS reads from the zero-padded table (merge to b64).
    v2f wf[7];
    #pragma unroll
    for (int s = 0; s < 7; ++s) {
        const int k = 4 * s + 2 * half;
        wf[s][0] = sGpad[GPAD + k - l16];
        wf[s][1] = sGpad[GPAD + k + 1 - l16];
    }

    // ---------- horizontal blur: 5 fields share each input load ----------
    // Row blocks cover Hb rows [0..15] and [12..27] (base 12 keeps the
    // transposed b128 spills 16B-aligned; rows 26,27 are killed by zero weights).
    v8f hacc[2][5];
    #pragma unroll
    for (int rb = 0; rb < 2; ++rb)
        #pragma unroll
        for (int f = 0; f < 5; ++f)
            hacc[rb][f] = (v8f){};

    #pragma unroll
    for (int rb = 0; rb < 2; ++rb) {
        const int r = ty + rb * 12 + l16;            // row in staged input
        #pragma unroll
        for (int s = 0; s < 7; ++s) {                // K = 28 in steps of 4
            const int c0 = tx + 4 * s + 2 * half;    // this lane's A pair: c0, c0+1
            const float p0 = sP[r][c0], p1 = sP[r][c0 + 1];
            const float t0 = sT[r][c0], t1 = sT[r][c0 + 1];
            v2f a0, a1, a2, a3, a4;
            a0[0] = p0;      a0[1] = p1;        // p
            a1[0] = t0;      a1[1] = t1;        // t
            a2[0] = p0 * p0; a2[1] = p1 * p1;   // p*p
            a3[0] = t0 * t0; a3[1] = t1 * t1;   // t*t
            a4[0] = p0 * t0; a4[1] = p1 * t1;   // p*t
            hacc[rb][0] = __builtin_amdgcn_wmma_f32_16x16x4_f32(
                              false, a0, false, wf[s], (short)0, hacc[rb][0], false, false);
            hacc[rb][1] = __builtin_amdgcn_wmma_f32_16x16x4_f32(
                              false, a1, false, wf[s], (short)0, hacc[rb][1], false, false);
            hacc[rb][2] = __builtin_amdgcn_wmma_f32_16x16x4_f32(
                              false, a2, false, wf[s], (short)0, hacc[rb][2], false, false);
            hacc[rb][3] = __builtin_amdgcn_wmma_f32_16x16x4_f32(
                              false, a3, false, wf[s], (short)0, hacc[rb][3], false, false);
            hacc[rb][4] = __builtin_amdgcn_wmma_f32_16x16x4_f32(
                              false, a4, false, wf[s], (short)0, hacc[rb][4], false, false);
        }
    }

    // ---------- per field: stage Hb (transposed) through LDS, vertical blur ----------
    // C/D layout: VGPR i -> row (base + i + 8*half), col l16. Column-major Hb
    // makes the 8 elements contiguous -> two b128 stores per block.
    // DS ops are in-order within a wave: scratch reuse across fields is safe.
    v8f res[5];
    #pragma unroll
    for (int f = 0; f < 5; ++f) {
        #pragma unroll
        for (int rb = 0; rb < 2; ++rb) {
            float* dst = &sHbT[wave][l16][rb * 12 + 8 * half];
            float4 lo, hi;
            lo.x = hacc[rb][f][0]; lo.y = hacc[rb][f][1];
            lo.z = hacc[rb][f][2]; lo.w = hacc[rb][f][3];
            hi.x = hacc[rb][f][4]; hi.y = hacc[rb][f][5];
            hi.z = hacc[rb][f][6]; hi.w = hacc[rb][f][7];
            *(float4*)(dst)     = lo;
            *(float4*)(dst + 4) = hi;
        }
        v8f accV = (v8f){};
        #pragma unroll
        for (int s = 0; s < 7; ++s) {
            const int k = 4 * s + 2 * half;
            v2f bf;                       // B[k][n] = Hb[k][n] = sHbT[n][k]
            bf[0] = sHbT[wave][l16][k];
            bf[1] = sHbT[wave][l16][k + 1];
            accV = __builtin_amdgcn_wmma_f32_16x16x4_f32(
                       false, wf[s], false, bf, (short)0, accV, false, false);
        }
        res[f] = accV;
    }

    // ---------- elementwise SSIM map + local sum ----------
    const float C1v = 0.0001f, C2v = 0.0009f, EPSV = 1e-8f;
    float lsum = 0.0f;
    #pragma unroll
    for (int i = 0; i < 8; ++i) {
        float mp  = res[0][i], mt  = res[1][i];
        float bp2 = res[2][i], bt2 = res[3][i], bpt = res[4][i];
        float mp2 = mp * mp, mt2 = mt * mt, mpt = mp * mt;
        float sp  = fmaxf(bp2 - mp2, 0.0f);
        float st  = fmaxf(bt2 - mt2, 0.0f);
        float spt = bpt - mpt;
        float num = (2.0f * mpt + C1v) * (2.0f * spt + C2v);
        float den = (mp2 + mt2 + C1v) * (sp + st + C2v);
        lsum += num / (den + EPSV);
    }

    // Deterministic reduction: wave shuffle -> LDS -> one partial per block.
    for (int off = 16; off > 0; off >>= 1)
        lsum += __shfl_down(lsum, off, 32);
    if (lane == 0) sRed[wave] = lsum;
    __syncthreads();
    if (tid == 0) {
        float t = 0.0f;
        for (int w2 = 0; w2 < NWAVE; ++w2) t += sRed[w2];
        int wg = (blockIdx.z * gridDim.y + blockIdx.y) * gridDim.x + blockIdx.x;
        partial[wg] = t;
    }
}

__global__ __launch_bounds__(256)
void ssim_finalize_kernel(const float* __restrict__ partial, int n,
                          float inv_count, float* __restrict__ out) {
    __shared__ float sm[256];
    float s = 0.0f;
    for (int i = threadIdx.x; i < n; i += 256) s += partial[i];
    sm[threadIdx.x] = s;
    __syncthreads();
    for (int st = 128; st > 0; st >>= 1) {
        if (threadIdx.x < st) sm[threadIdx.x] += sm[threadIdx.x + st];
        __syncthreads();
    }
    if (threadIdx.x == 0) out[0] = 1.0f - sm[0] * inv_count;
}

extern "C" void kernel_launch(void* const* d_in, const int* in_sizes, int n_in,
                              void* d_out, int out_size, void* d_ws, size_t ws_size,
                              hipStream_t stream) {
    const float* pred = (const float*)d_in[0];
    const float* tgt  = (const float*)d_in[1];
    const float* win  = (const float*)d_in[2];
    float* out = (float*)d_out;
    float* partial = (float*)d_ws;

    const int B = 32, H = 512, W = 512;
    dim3 grid(W / TILE_W, H / TILE_H, B);   // 8 x 16 x 32 = 4096 blocks
    ssim_main_kernel<<<grid, 256, 0, stream>>>(pred, tgt, win, partial, H, W);

    const int nPartials = (int)(grid.x * grid.y * grid.z);
    const float invCount = 1.0f / ((float)B * (float)H * (float)W);
    ssim_finalize_kernel<<<1, 256, 0, stream>>>(partial, nPartials, invCount, out);
}